// SparseMoE_17944373363073
// MI455X (gfx1250) — compile-verified
//
#include <hip/hip_runtime.h>
#include <math.h>

#define T_TOK    8192
#define DM       1024
#define DH       4096
#define N_EXP    8

typedef __attribute__((ext_vector_type(2))) float v2f;
typedef __attribute__((ext_vector_type(4))) float v4f;
typedef __attribute__((ext_vector_type(8))) float v8f;

// ---- workspace layout (bytes) ----
// ybuf : T*2*DM floats = 64 MiB      @ 0
// etok : N_EXP*T ints  = 256 KiB     @ 67108864
// ew   : N_EXP*T floats= 256 KiB     @ 67371008
// cnt  : N_EXP ints                  @ 67633152
#define WS_ETOK_OFF 67108864u
#define WS_EW_OFF   67371008u
#define WS_CNT_OFF  67633152u

__global__ void zero_counts_kernel(int* counts) {
    if (threadIdx.x < N_EXP) counts[threadIdx.x] = 0;
}

__global__ __launch_bounds__(256)
void gate_kernel(const float* __restrict__ x, const float* __restrict__ gw,
                 int* __restrict__ counts, int* __restrict__ etok,
                 float* __restrict__ ew) {
    __shared__ float gws[N_EXP * DM];            // 32 KB
    for (int i = threadIdx.x; i < N_EXP * DM; i += blockDim.x) gws[i] = gw[i];
    __syncthreads();

    const int t = blockIdx.x * blockDim.x + threadIdx.x;
    const float* xp = x + (size_t)t * DM;
    float lg[N_EXP];
#pragma unroll
    for (int e = 0; e < N_EXP; ++e) lg[e] = 0.f;

    for (int k = 0; k < DM; k += 4) {
        v4f xv = *(const v4f*)(xp + k);
#pragma unroll
        for (int e = 0; e < N_EXP; ++e) {
            const float* g = &gws[e * DM + k];
            lg[e] += xv.x * g[0] + xv.y * g[1] + xv.z * g[2] + xv.w * g[3];
        }
    }
    // top-2 with lowest-index tie-break (matches jax.lax.top_k)
    int i0 = 0;
#pragma unroll
    for (int e = 1; e < N_EXP; ++e) if (lg[e] > lg[i0]) i0 = e;
    int i1 = (i0 == 0) ? 1 : 0;
#pragma unroll
    for (int e = 0; e < N_EXP; ++e) if (e != i1 && e != i0 && lg[e] > lg[i1]) i1 = e;

    // softmax over the two (lg[i0] >= lg[i1], numerically safe)
    float r0 = 1.0f / (1.0f + expf(lg[i1] - lg[i0]));
    float r1 = 1.0f - r0;

    int p0 = atomicAdd(&counts[i0], 1);
    etok[i0 * T_TOK + p0] = (t << 1) | 0;
    ew[i0 * T_TOK + p0]   = r0;
    int p1 = atomicAdd(&counts[i1], 1);
    etok[i1 * T_TOK + p1] = (t << 1) | 1;
    ew[i1 * T_TOK + p1]   = r1;
}

// LDS geometry (floats)
#define XS_STRIDE 1028                       // 16 rows x 1024 + pad
#define HS_STRIDE 264                        // 16 rows x 256 chunk + pad
#define XS_ELEMS  (16 * XS_STRIDE)
#define HS_ELEMS  (16 * HS_STRIDE)
#define SMEM_BYTES ((XS_ELEMS + HS_ELEMS) * 4 + 16 * 4 + 16 * 4)

__global__ __launch_bounds__(256)
void moe_ffn_kernel(const float* __restrict__ x,
                    const float* __restrict__ w1, const float* __restrict__ b1,
                    const float* __restrict__ w2, const float* __restrict__ b2,
                    const int* __restrict__ counts,
                    const int* __restrict__ etok, const float* __restrict__ ew,
                    float* __restrict__ ybuf) {
    const int e        = blockIdx.y;
    const int cnt      = counts[e];
    const int tileBase = blockIdx.x * 16;
    if (tileBase >= cnt) return;             // uniform exit, EXEC stays full

    extern __shared__ char smem[];
    float* Xs   = (float*)smem;              // [16][XS_STRIDE]
    float* Hs   = Xs + XS_ELEMS;             // [16][HS_STRIDE]
    int*   sTok = (int*)(Hs + HS_ELEMS);     // [16]
    float* sW   = (float*)(sTok + 16);       // [16]

    const int tid  = threadIdx.x;
    const int lane = tid & 31;
    const int wv   = tid >> 5;               // wave 0..7
    const int ln   = lane & 15;
    const int hi   = lane >> 4;              // K-half select per ISA A/B layout

    if (tid < 16) {
        int idx = tileBase + tid;
        if (idx < cnt) { sTok[tid] = etok[e * T_TOK + idx]; sW[tid] = ew[e * T_TOK + idx]; }
        else           { sTok[tid] = -1;                    sW[tid] = 0.f; }
    }
    __syncthreads();

    // gather 16 token rows of X into LDS (rows for padded slots are dummies,
    // harmless: GEMM rows are independent and their stores are masked off)
    {
        int m = tid >> 4, cb = (tid & 15) * 64;
        int ts = sTok[m];
        const float* xr = x + (size_t)(ts < 0 ? 0 : (ts >> 1)) * DM + cb;
        float* xd = Xs + m * XS_STRIDE + cb;
#pragma unroll
        for (int i = 0; i < 64; i += 4) *(v4f*)(xd + i) = *(const v4f*)(xr + i);
    }
    __syncthreads();

    const float* w1p = w1 + (size_t)e * DM * DH;
    const float* w2p = w2 + (size_t)e * DH * DM;

    v8f acc[8];                              // this wave's Y[16 x 128] slice
#pragma unroll
    for (int i = 0; i < 8; ++i) { v8f z = {}; acc[i] = z; }

#pragma unroll 1
    for (int c0 = 0; c0 < DH; c0 += 256) {
        // ---- step 1: H chunk = relu(X @ W1 + b1), wave computes 2 tiles ----
#pragma unroll 1
        for (int t2 = 0; t2 < 2; ++t2) {
            const int nloc = wv * 32 + t2 * 16;
            const int n0   = c0 + nloc;
            v8f h = {};
#pragma unroll 4
            for (int k0 = 0; k0 < DM; k0 += 4) {
                v2f a = *(const v2f*)(Xs + ln * XS_STRIDE + k0 + 2 * hi);
                const float* bp = w1p + (size_t)(k0 + 2 * hi) * DH + n0 + ln;
                v2f b; b.x = bp[0]; b.y = bp[DH];
                h = __builtin_amdgcn_wmma_f32_16x16x4_f32(
                        false, a, false, b, (short)0, h, false, false);
            }
            float bias = b1[e * DH + n0 + ln];
            float* hd = Hs + (8 * hi) * HS_STRIDE + nloc + ln;
#pragma unroll
            for (int r = 0; r < 8; ++r) {
                float v = h[r] + bias;
                hd[r * HS_STRIDE] = v > 0.f ? v : 0.f;
            }
        }
        __syncthreads();
        // ---- step 2: Y += H_chunk @ W2_chunk (wave owns cols wv*128..+127) ----
#pragma unroll 1
        for (int kh = 0; kh < 256; kh += 4) {
            v2f a = *(const v2f*)(Hs + ln * HS_STRIDE + kh + 2 * hi);
            const float* bp = w2p + (size_t)(c0 + kh + 2 * hi) * DM + wv * 128 + ln;
#pragma unroll
            for (int ty = 0; ty < 8; ++ty) {
                v2f b; b.x = bp[ty * 16]; b.y = bp[ty * 16 + DM];
                acc[ty] = __builtin_amdgcn_wmma_f32_16x16x4_f32(
                              false, a, false, b, (short)0, acc[ty], false, false);
            }
        }
        __syncthreads();
    }

    // ---- epilogue: (+b2) * routing weight -> per-(token,slot) buffer ----
#pragma unroll 1
    for (int ty = 0; ty < 8; ++ty) {
        int n = wv * 128 + ty * 16 + ln;
        float b2v = b2[e * DM + n];
#pragma unroll
        for (int r = 0; r < 8; ++r) {
            int m = r + 8 * hi;
            if (tileBase + m < cnt) {
                int ts = sTok[m];
                ybuf[(size_t)ts * DM + n] = (acc[ty][r] + b2v) * sW[m];
            }
        }
    }
}

__global__ __launch_bounds__(256)
void reduce_kernel(const float* __restrict__ ybuf, float* __restrict__ out) {
    size_t i = ((size_t)blockIdx.x * blockDim.x + threadIdx.x) * 4;
    size_t t = i >> 10;          // /DM
    size_t d = i & (DM - 1);
    v4f y0 = *(const v4f*)(ybuf + t * 2 * DM + d);
    v4f y1 = *(const v4f*)(ybuf + t * 2 * DM + DM + d);
    *(v4f*)(out + i) = y0 + y1;
}

extern "C" void kernel_launch(void* const* d_in, const int* in_sizes, int n_in,
                              void* d_out, int out_size, void* d_ws, size_t ws_size,
                              hipStream_t stream) {
    const float* x      = (const float*)d_in[0];
    const float* gate_w = (const float*)d_in[1];
    const float* w1     = (const float*)d_in[2];
    const float* b1     = (const float*)d_in[3];
    const float* w2     = (const float*)d_in[4];
    const float* b2     = (const float*)d_in[5];
    float* out = (float*)d_out;

    char* ws = (char*)d_ws;
    float* ybuf  = (float*)ws;
    int*   etok  = (int*)(ws + WS_ETOK_OFF);
    float* ew    = (float*)(ws + WS_EW_OFF);
    int*   cntrs = (int*)(ws + WS_CNT_OFF);

    zero_counts_kernel<<<1, 32, 0, stream>>>(cntrs);
    gate_kernel<<<T_TOK / 256, 256, 0, stream>>>(x, gate_w, cntrs, etok, ew);
    dim3 grid(T_TOK / 16, N_EXP);
    moe_ffn_kernel<<<grid, 256, SMEM_BYTES, stream>>>(x, w1, b1, w2, b2,
                                                      cntrs, etok, ew, ybuf);
    reduce_kernel<<<(T_TOK * DM / 4) / 256, 256, 0, stream>>>(ybuf, out);
}